// GCN_78675210928386
// MI455X (gfx1250) — compile-verified
//
#include <hip/hip_runtime.h>
#include <math.h>
#include <float.h>

// ---------------------------------------------------------------------------
// GCN on MI455X (gfx1250). fp32 end-to-end to match the reference.
// Dense node transforms use V_WMMA_F32_16X16X4_F32 (wave32 WMMA).
// Edge aggregation uses fp32 global atomics (L2-resident working set).
//
// Invariant maintained by the launcher: nrows (=NN=50000) is a multiple of 16
// and the GEMM grid is exactly nrows/16 tiles, so the WMMA kernel has NO
// bounds guards -> straight-line load/WMMA/store with uniform EXEC.
// ---------------------------------------------------------------------------

typedef __attribute__((ext_vector_type(2))) float v2f;
typedef __attribute__((ext_vector_type(8))) float v8f;

#define NN 50000
#define EE 1200000
#define DD 64
#define GG 1024

static_assert(NN % 16 == 0, "WMMA GEMM requires row count multiple of 16");

// ---------------------------- small utility kernels ------------------------

__global__ void fill_f32_kernel(float* __restrict__ p, float v, int n) {
  int i = blockIdx.x * blockDim.x + threadIdx.x;
  if (i < n) p[i] = v;
}

__global__ void degree_kernel(const int* __restrict__ dst, float* __restrict__ deg, int e) {
  int i = blockIdx.x * blockDim.x + threadIdx.x;
  if (i < e) atomicAdd(&deg[dst[i]], 1.0f);
}

__global__ void dinv_kernel(const float* __restrict__ deg, float* __restrict__ dinv, int n) {
  int i = blockIdx.x * blockDim.x + threadIdx.x;
  if (i < n) dinv[i] = rsqrtf(deg[i]);
}

// pad x [N,9] -> [N,16] (zero-filled K tail) so the WMMA GEMM can run K=16
__global__ void pad_x_kernel(const float* __restrict__ x, float* __restrict__ hpad, int n) {
  int i = blockIdx.x * blockDim.x + threadIdx.x;
  if (i >= n * 16) return;
  int r = i >> 4, c = i & 15;
  hpad[i] = (c < 9) ? x[r * 9 + c] : 0.0f;
}

// pad W0 [9,64] -> [16,64]
__global__ void pad_w0_kernel(const float* __restrict__ W0, float* __restrict__ w0pad) {
  int i = blockIdx.x * blockDim.x + threadIdx.x;
  if (i >= 16 * 64) return;
  int r = i >> 6, c = i & 63;
  w0pad[i] = (r < 9) ? W0[r * 64 + c] : 0.0f;
}

// ------------------------------- WMMA GEMM ---------------------------------
// out[N,64] = A[N,KDIM] @ W[KDIM,64]. One 16-row tile per block; 4 waves per
// block each own one 16-column tile. Full-K accumulation via chained
// V_WMMA_F32_16X16X4_F32 (KDIM/4 steps, fully unrolled). No guards: grid
// geometry guarantees every tile is full, EXEC stays all-ones throughout.
//
// Per the CDNA5 VGPR layouts (CDNA5 ISA):
//  A 16x4 f32 : lane L holds row M=L%16; VGPRs {v0,v1} = K = (L/16)*2 + {0,1}
//  B 4x16 f32 : lane L holds col N=L%16; VGPRs {v0,v1} = K = (L/16)*2 + {0,1}
//  C/D 16x16  : lane L, VGPR v -> M = v + 8*(L/16), N = L%16
template <int KDIM>
__global__ void gemm_wmma_kernel(const float* __restrict__ A,
                                 const float* __restrict__ W,
                                 float* __restrict__ out) {
  const int wave = threadIdx.x >> 5;        // 0..3 -> column tile
  const int lane = threadIdx.x & 31;
  const int l15  = lane & 15;
  const int half = lane >> 4;               // 0 or 1
  const int m0   = blockIdx.x * 16;
  const int n0   = wave * 16;
  const int row  = m0 + l15;
  const int col  = n0 + l15;

  const float* __restrict__ arow = A + (size_t)row * KDIM + half * 2;
  const float* __restrict__ bcol = W + (size_t)half * 2 * 64 + col;

  v8f c = {};
#pragma unroll
  for (int k = 0; k < KDIM; k += 4) {
    v2f a, b;
    a.x = arow[k + 0];
    a.y = arow[k + 1];
    b.x = bcol[(size_t)k * 64];
    b.y = bcol[(size_t)k * 64 + 64];
    // 8 args: (neg_a, A, neg_b, B, c_mod, C, reuse_a, reuse_b)
    c = __builtin_amdgcn_wmma_f32_16x16x4_f32(false, a, false, b, (short)0, c,
                                              false, false);
  }

  // straight-line store burst: for each v, lanes 0-15 cover one contiguous
  // 64B row segment, lanes 16-31 the row 8 below it.
  float* __restrict__ orow = out + (size_t)(m0 + half * 8) * 64 + col;
#pragma unroll
  for (int v = 0; v < 8; ++v) {
    orow[(size_t)v * 64] = c[v];
  }
}

// --------------------------- edge scatter-add ------------------------------
// agg[dst] += hW[src] * dinv[src]*dinv[dst]; 16 lanes per edge, float4 each.
__global__ void edge_agg_kernel(const float* __restrict__ hW,
                                const int* __restrict__ src,
                                const int* __restrict__ dst,
                                const float* __restrict__ dinv,
                                float* __restrict__ agg, int e) {
  int idx = blockIdx.x * blockDim.x + threadIdx.x;
  if (idx >= e * 16) return;
  const int ei = idx >> 4;
  const int fg = (idx & 15) * 4;
  const int s = src[ei];
  const int d = dst[ei];
  const float norm = dinv[s] * dinv[d];
  const float4 hv = *(const float4*)(hW + (size_t)s * DD + fg);
  float* ap = agg + (size_t)d * DD + fg;
  atomicAdd(ap + 0, hv.x * norm);
  atomicAdd(ap + 1, hv.y * norm);
  atomicAdd(ap + 2, hv.z * norm);
  atomicAdd(ap + 3, hv.w * norm);
}

// self-loop + bias + tanh:  h = tanh(agg + hW*dinv^2 + b)
__global__ void layer_finish_kernel(const float* __restrict__ agg,
                                    const float* __restrict__ hW,
                                    const float* __restrict__ dinv,
                                    const float* __restrict__ b,
                                    float* __restrict__ hout, int n) {
  int i = blockIdx.x * blockDim.x + threadIdx.x;
  if (i >= n * DD) return;
  const int row = i >> 6;
  const int f = i & 63;
  const float di = dinv[row];
  hout[i] = tanhf(agg[i] + hW[i] * di * di + b[f]);
}

// ------------------------------- pooling -----------------------------------

__device__ inline void atomicMaxF32(float* addr, float val) {
  if (val >= 0.0f)
    atomicMax((int*)addr, __float_as_int(val));
  else
    atomicMin((unsigned int*)addr, __float_as_uint(val));
}

__global__ void pool_acc_kernel(const float* __restrict__ h,
                                const int* __restrict__ batch,
                                float* __restrict__ gmax,
                                float* __restrict__ gsum,
                                float* __restrict__ cnt, int n) {
  int i = blockIdx.x * blockDim.x + threadIdx.x;
  if (i >= n * DD) return;
  const int row = i >> 6;
  const int f = i & 63;
  const int g = batch[row];
  const float v = h[i];
  atomicAdd(&gsum[(size_t)g * DD + f], v);
  atomicMaxF32(&gmax[(size_t)g * DD + f], v);
  if (f == 0) atomicAdd(&cnt[g], 1.0f);
}

// hidden[g] = concat(gmax_guarded, gsum/max(cnt,1)); hidden written to d_out+G
__global__ void pool_finish_kernel(const float* __restrict__ gmax,
                                   const float* __restrict__ gsum,
                                   const float* __restrict__ cnt,
                                   float* __restrict__ hidden, int g_) {
  int i = blockIdx.x * blockDim.x + threadIdx.x;
  if (i >= g_ * DD) return;
  const int g = i >> 6;
  const int f = i & 63;
  const float c = cnt[g];
  float mx = gmax[(size_t)g * DD + f];
  if (c == 0.0f) mx = 0.0f;  // matches jnp.where(isfinite, ., 0) for empty segs
  hidden[(size_t)g * (2 * DD) + f] = mx;
  hidden[(size_t)g * (2 * DD) + DD + f] = gsum[(size_t)g * DD + f] / fmaxf(c, 1.0f);
}

// out[g] = hidden[g,:] . Wout + bout
__global__ void head_kernel(const float* __restrict__ hidden,
                            const float* __restrict__ Wout,
                            const float* __restrict__ bout,
                            float* __restrict__ out, int g_) {
  int g = blockIdx.x * blockDim.x + threadIdx.x;
  if (g >= g_) return;
  float acc = bout[0];
#pragma unroll 8
  for (int k = 0; k < 2 * DD; ++k) acc += hidden[(size_t)g * (2 * DD) + k] * Wout[k];
  out[g] = acc;
}

// ------------------------------- launcher ----------------------------------

extern "C" void kernel_launch(void* const* d_in, const int* in_sizes, int n_in,
                              void* d_out, int out_size, void* d_ws, size_t ws_size,
                              hipStream_t stream) {
  (void)in_sizes; (void)n_in; (void)out_size; (void)ws_size;

  const float* x    = (const float*)d_in[0];
  const int*   eidx = (const int*)d_in[1];       // [2,E]
  const int*   batch = (const int*)d_in[2];      // [N]
  const float* W0 = (const float*)d_in[3];
  const float* b0 = (const float*)d_in[4];
  const float* W1 = (const float*)d_in[5];
  const float* b1 = (const float*)d_in[6];
  const float* W2 = (const float*)d_in[7];
  const float* b2 = (const float*)d_in[8];
  const float* W3 = (const float*)d_in[9];
  const float* b3 = (const float*)d_in[10];
  const float* Wout = (const float*)d_in[11];
  const float* bout = (const float*)d_in[12];

  const int* src = eidx;
  const int* dst = eidx + EE;

  float* outp   = (float*)d_out;        // [G]
  float* hidden = (float*)d_out + GG;   // [G, 128]

  // workspace layout (floats)
  float* ws = (float*)d_ws;
  size_t o = 0;
  float* deg   = ws + o; o += NN;
  float* dinv  = ws + o; o += NN;
  float* hpad  = ws + o; o += (size_t)NN * 16;
  float* w0pad = ws + o; o += 16 * 64;
  float* hW    = ws + o; o += (size_t)NN * DD;
  float* h     = ws + o; o += (size_t)NN * DD;
  float* agg   = ws + o; o += (size_t)NN * DD;
  float* gmax  = ws + o; o += (size_t)GG * DD;
  float* gsum  = ws + o; o += (size_t)GG * DD;
  float* cnt   = ws + o; o += GG;

  const int T = 256;
  auto nb = [](long long n, int t) { return (int)((n + t - 1) / t); };

  // --- degrees & normalization ---
  fill_f32_kernel<<<nb(NN, T), T, 0, stream>>>(deg, 1.0f, NN);  // self-loop
  degree_kernel<<<nb(EE, T), T, 0, stream>>>(dst, deg, EE);
  dinv_kernel<<<nb(NN, T), T, 0, stream>>>(deg, dinv, NN);

  // --- padded inputs for layer 0 ---
  pad_x_kernel<<<nb((long long)NN * 16, T), T, 0, stream>>>(x, hpad, NN);
  pad_w0_kernel<<<nb(16 * 64, T), T, 0, stream>>>(W0, w0pad);

  const int mtiles = NN / 16;           // exact: 3125 full tiles, no guards
  const float* Wl[4] = {w0pad, W1, W2, W3};
  const float* bl[4] = {b0, b1, b2, b3};

  for (int l = 0; l < 4; ++l) {
    // dense transform: hW = A @ W (WMMA f32)
    if (l == 0)
      gemm_wmma_kernel<16><<<mtiles, 128, 0, stream>>>(hpad, Wl[0], hW);
    else
      gemm_wmma_kernel<64><<<mtiles, 128, 0, stream>>>(h, Wl[l], hW);

    // zero agg, scatter edges, fuse self-loop+bias+tanh
    fill_f32_kernel<<<nb((long long)NN * DD, T), T, 0, stream>>>(agg, 0.0f, NN * DD);
    edge_agg_kernel<<<nb((long long)EE * 16, T), T, 0, stream>>>(hW, src, dst, dinv, agg, EE);
    layer_finish_kernel<<<nb((long long)NN * DD, T), T, 0, stream>>>(agg, hW, dinv, bl[l], h, NN);
  }

  // --- pooling ---
  fill_f32_kernel<<<nb((long long)GG * DD, T), T, 0, stream>>>(gmax, -FLT_MAX, GG * DD);
  fill_f32_kernel<<<nb((long long)GG * DD, T), T, 0, stream>>>(gsum, 0.0f, GG * DD);
  fill_f32_kernel<<<nb(GG, T), T, 0, stream>>>(cnt, 0.0f, GG);
  pool_acc_kernel<<<nb((long long)NN * DD, T), T, 0, stream>>>(h, batch, gmax, gsum, cnt, NN);
  pool_finish_kernel<<<nb((long long)GG * DD, T), T, 0, stream>>>(gmax, gsum, cnt, hidden, GG);

  // --- readout head ---
  head_kernel<<<nb(GG, 128), 128, 0, stream>>>(hidden, Wout, bout, outp, GG);
}